// GATEncoderP_64785286693464
// MI455X (gfx1250) — compile-verified
//
#include <hip/hip_runtime.h>
#include <math.h>

#define N_NODES 50000
#define N_EDGES 800000
#define EP (N_EDGES + N_NODES)   // edges + self loops
#define IN_CH 128
#define HID 64
#define HEADS 4
#define D1 (HEADS * HID)         // 256

typedef __attribute__((ext_vector_type(2))) float v2f;
typedef __attribute__((ext_vector_type(8))) float v8f;

// ---------- monotonic float<->uint key for atomic segment-max ----------
__device__ __forceinline__ unsigned fkey(float f) {
    unsigned u = __float_as_uint(f);
    return (u & 0x80000000u) ? ~u : (u | 0x80000000u);
}
__device__ __forceinline__ float fdec(unsigned k) {
    unsigned u = (k & 0x80000000u) ? (k & 0x7FFFFFFFu) : ~k;
    return __uint_as_float(u);
}

__device__ __forceinline__ void edge_sd(const int* __restrict__ ei, int e, int& s, int& d) {
    if (e < N_EDGES) { s = ei[e]; d = ei[N_EDGES + e]; }
    else             { s = e - N_EDGES; d = s; }      // appended self-loop
}

// ---------------------------------------------------------------------
// Dense projection: C[M x NC] = A[M x K] * B[K x NC], f32 via
// V_WMMA_F32_16X16X4_F32. One wave -> one 16x16 tile; 4 waves/block share
// a 16xK A-tile staged through LDS. M (=50000) is a multiple of 16.
// ---------------------------------------------------------------------
__global__ void gemm_wmma_f32(const float* __restrict__ A,
                              const float* __restrict__ B,
                              float* __restrict__ Cout,
                              int K, int NC) {
    __shared__ float As[16 * 256];                 // up to K=256 (16 KB of 320 KB)
    const int m0   = blockIdx.x * 16;
    const int tid  = threadIdx.x;
    const int wave = tid >> 5;
    const int lane = tid & 31;

    for (int i = tid; i < 16 * K; i += blockDim.x)
        As[i] = A[(size_t)(m0 + i / K) * K + (i % K)];
    __syncthreads();

    const int n0 = (blockIdx.y * 4 + wave) * 16;
    if (n0 >= NC) return;

    const int half = lane >> 4;                    // 0: lanes 0-15, 1: lanes 16-31
    const int rl   = lane & 15;

    v8f c = {0.f, 0.f, 0.f, 0.f, 0.f, 0.f, 0.f, 0.f};
    for (int k0 = 0; k0 < K; k0 += 4) {
        const int kk = k0 + (half << 1);           // K{0,1} | K{2,3} split per ISA layout
        v2f a, b;
        a[0] = As[rl * K + kk];
        a[1] = As[rl * K + kk + 1];
        b[0] = B[(size_t)kk * NC + n0 + rl];       // row K=kk of B, N striped on lanes
        b[1] = B[(size_t)(kk + 1) * NC + n0 + rl];
        c = __builtin_amdgcn_wmma_f32_16x16x4_f32(false, a, false, b,
                                                  (short)0, c, false, false);
    }

    const int rbase = m0 + (half << 3);            // VGPR i -> rows i and i+8
#pragma unroll
    for (int i = 0; i < 8; ++i)
        Cout[(size_t)(rbase + i) * NC + n0 + rl] = c[i];
}

// ---------------------------------------------------------------------
// Per-(node, head) attention coefficients; also init segment max/sum.
// ---------------------------------------------------------------------
__global__ void attn_coef(const float* __restrict__ h,
                          const float* __restrict__ a_src,
                          const float* __restrict__ a_dst,
                          float* __restrict__ asrc, float* __restrict__ adst,
                          unsigned* __restrict__ maxk, float* __restrict__ sums,
                          int H, int C) {
    int gid = blockIdx.x * blockDim.x + threadIdx.x;
    if (gid >= N_NODES * H) return;
    int n = gid / H, hh = gid - n * H;
    const float* hp = h + (size_t)n * H * C + (size_t)hh * C;
    const float* s  = a_src + hh * C;
    const float* d  = a_dst + hh * C;
    float s1 = 0.f, s2 = 0.f;
    for (int c = 0; c < C; ++c) { float v = hp[c]; s1 += v * s[c]; s2 += v * d[c]; }
    asrc[gid] = s1;
    adst[gid] = s2;
    maxk[gid] = fkey(-INFINITY);
    sums[gid] = 0.f;
}

// ---------------------------------------------------------------------
// Edge phase 1: leaky-relu logit + segment max via uint-keyed atomicMax.
// ---------------------------------------------------------------------
__global__ void edge_logit_max(const int* __restrict__ ei,
                               const float* __restrict__ asrc,
                               const float* __restrict__ adst,
                               float* __restrict__ logit,
                               unsigned* __restrict__ maxk, int H) {
    int gid = blockIdx.x * blockDim.x + threadIdx.x;
    if (gid >= EP * H) return;
    int e = gid / H, hh = gid - e * H;
    int s, d; edge_sd(ei, e, s, d);
    float l = asrc[(size_t)s * H + hh] + adst[(size_t)d * H + hh];
    l = l > 0.f ? l : 0.2f * l;                    // GAT negative_slope
    logit[gid] = l;
    atomicMax(&maxk[(size_t)d * H + hh], fkey(l));
}

// ---------------------------------------------------------------------
// Edge phase 2: exp(logit - max[dst]) in place + segment sum.
// ---------------------------------------------------------------------
__global__ void edge_exp_sum(const int* __restrict__ ei,
                             float* __restrict__ w,
                             const unsigned* __restrict__ maxk,
                             float* __restrict__ sums, int H) {
    int gid = blockIdx.x * blockDim.x + threadIdx.x;
    if (gid >= EP * H) return;
    int e = gid / H, hh = gid - e * H;
    int s, d; edge_sd(ei, e, s, d);
    float m  = fdec(maxk[(size_t)d * H + hh]);
    float ex = expf(w[gid] - m);
    w[gid] = ex;
    atomicAdd(&sums[(size_t)d * H + hh], ex);
}

// ---------------------------------------------------------------------
// out[n][c] = bias[c]  (bias folded into aggregation base).
// ---------------------------------------------------------------------
__global__ void init_bias(float* __restrict__ out, const float* __restrict__ bias, int D) {
    long long gid = (long long)blockIdx.x * blockDim.x + threadIdx.x;
    if (gid >= (long long)N_NODES * D) return;
    out[gid] = bias[gid % D];
}

// ---------------------------------------------------------------------
// Edge phase 3: normalized weighted scatter. Thread = (edge, feature);
// consecutive threads hit consecutive channels -> coalesced gather and
// coalesced GLOBAL_ATOMIC_ADD_F32 (no return).
// ---------------------------------------------------------------------
__global__ void edge_scatter(const int* __restrict__ ei,
                             const float* __restrict__ h,
                             const float* __restrict__ w,
                             const float* __restrict__ sums,
                             float* __restrict__ out, int H, int C) {
    long long gid = (long long)blockIdx.x * blockDim.x + threadIdx.x;
    const int D = H * C;
    if (gid >= (long long)EP * D) return;
    int e = (int)(gid / D);
    int r = (int)(gid - (long long)e * D);
    int hh = r / C;
    int s, d; edge_sd(ei, e, s, d);
    float wn = w[(size_t)e * H + hh] / (sums[(size_t)d * H + hh] + 1e-16f);
    atomicAdd(&out[(size_t)d * D + r], h[(size_t)s * D + r] * wn);
}

// ---------------------------------------------------------------------
// LayerNorm(256) + LeakyReLU(0.01), one wave32 per node, 8 elems/lane.
// ---------------------------------------------------------------------
__global__ void ln_lrelu(float* __restrict__ x,
                         const float* __restrict__ gamma,
                         const float* __restrict__ beta) {
    int node = blockIdx.x * (blockDim.x >> 5) + (threadIdx.x >> 5);
    int lane = threadIdx.x & 31;
    if (node >= N_NODES) return;
    float* row = x + (size_t)node * D1;
    float v[8];
    float s = 0.f;
#pragma unroll
    for (int i = 0; i < 8; ++i) { v[i] = row[lane + 32 * i]; s += v[i]; }
#pragma unroll
    for (int m = 16; m > 0; m >>= 1) s += __shfl_xor(s, m, 32);
    float mu = s * (1.f / 256.f);
    float q = 0.f;
#pragma unroll
    for (int i = 0; i < 8; ++i) { float t = v[i] - mu; q += t * t; }
#pragma unroll
    for (int m = 16; m > 0; m >>= 1) q += __shfl_xor(q, m, 32);
    float rs = rsqrtf(q * (1.f / 256.f) + 1e-5f);
#pragma unroll
    for (int i = 0; i < 8; ++i) {
        int c = lane + 32 * i;
        float y = (v[i] - mu) * rs * gamma[c] + beta[c];
        row[c] = y > 0.f ? y : 0.01f * y;
    }
}

// ---------------------------------------------------------------------
extern "C" void kernel_launch(void* const* d_in, const int* in_sizes, int n_in,
                              void* d_out, int out_size, void* d_ws, size_t ws_size,
                              hipStream_t stream) {
    const float* x     = (const float*)d_in[0];
    const int*   ei    = (const int*)  d_in[1];
    const float* W1    = (const float*)d_in[2];
    const float* as1   = (const float*)d_in[3];
    const float* ad1   = (const float*)d_in[4];
    const float* b1    = (const float*)d_in[5];
    const float* gamma = (const float*)d_in[6];
    const float* beta  = (const float*)d_in[7];
    const float* W2    = (const float*)d_in[8];
    const float* as2   = (const float*)d_in[9];
    const float* ad2   = (const float*)d_in[10];
    const float* b2    = (const float*)d_in[11];
    float* out = (float*)d_out;

    float* ws = (float*)d_ws;
    float*    h1    = ws;               ws += (size_t)N_NODES * D1;
    float*    agg1  = ws;               ws += (size_t)N_NODES * D1;
    float*    h2    = ws;               ws += (size_t)N_NODES * HID;
    float*    asrc1 = ws;               ws += (size_t)N_NODES * HEADS;
    float*    adst1 = ws;               ws += (size_t)N_NODES * HEADS;
    unsigned* maxk1 = (unsigned*)ws;    ws += (size_t)N_NODES * HEADS;
    float*    sum1  = ws;               ws += (size_t)N_NODES * HEADS;
    float*    w1    = ws;               ws += (size_t)EP * HEADS;
    float*    asrc2 = ws;               ws += N_NODES;
    float*    adst2 = ws;               ws += N_NODES;
    unsigned* maxk2 = (unsigned*)ws;    ws += N_NODES;
    float*    sum2  = ws;               ws += N_NODES;
    float*    w2    = ws;               ws += (size_t)EP;

    // ---------------- Layer 1 (H=4, C=64) ----------------
    gemm_wmma_f32<<<dim3(N_NODES / 16, D1 / 64), 128, 0, stream>>>(x, W1, h1, IN_CH, D1);
    attn_coef<<<(N_NODES * HEADS + 255) / 256, 256, 0, stream>>>(
        h1, as1, ad1, asrc1, adst1, maxk1, sum1, HEADS, HID);
    edge_logit_max<<<((size_t)EP * HEADS + 255) / 256, 256, 0, stream>>>(
        ei, asrc1, adst1, w1, maxk1, HEADS);
    edge_exp_sum<<<((size_t)EP * HEADS + 255) / 256, 256, 0, stream>>>(
        ei, w1, maxk1, sum1, HEADS);
    init_bias<<<((size_t)N_NODES * D1 + 255) / 256, 256, 0, stream>>>(agg1, b1, D1);
    edge_scatter<<<((size_t)EP * D1 + 255) / 256, 256, 0, stream>>>(
        ei, h1, w1, sum1, agg1, HEADS, HID);
    ln_lrelu<<<(N_NODES + 7) / 8, 256, 0, stream>>>(agg1, gamma, beta);

    // ---------------- Layer 2 (H=1, C=64) ----------------
    gemm_wmma_f32<<<dim3(N_NODES / 16, 1), 128, 0, stream>>>(agg1, W2, h2, D1, HID);
    attn_coef<<<(N_NODES + 255) / 256, 256, 0, stream>>>(
        h2, as2, ad2, asrc2, adst2, maxk2, sum2, 1, HID);
    edge_logit_max<<<(EP + 255) / 256, 256, 0, stream>>>(ei, asrc2, adst2, w2, maxk2, 1);
    edge_exp_sum<<<(EP + 255) / 256, 256, 0, stream>>>(ei, w2, maxk2, sum2, 1);
    init_bias<<<((size_t)N_NODES * HID + 255) / 256, 256, 0, stream>>>(out, b2, HID);
    edge_scatter<<<((size_t)EP * HID + 255) / 256, 256, 0, stream>>>(
        ei, h2, w2, sum2, out, 1, HID);
}